// QCNN_54863912239915
// MI455X (gfx1250) — compile-verified
//
#include <hip/hip_runtime.h>
#include <math.h>

typedef __attribute__((ext_vector_type(2))) float v2f;
typedef __attribute__((ext_vector_type(8))) float v8f;

static __device__ __forceinline__ float2 cmul(float2 a, float2 b) {
    return make_float2(a.x * b.x - a.y * b.y, a.x * b.y + a.y * b.x);
}
static __device__ __forceinline__ float2 shflx2(float2 v, int m) {
    return make_float2(__shfl_xor(v.x, m, 32), __shfl_xor(v.y, m, 32));
}

static __device__ __forceinline__ v8f wmma_f32(v2f a, v2f b, v8f c) {
    // D = A(16x4) x B(4x16) + C(16x16), full fp32
    return __builtin_amdgcn_wmma_f32_16x16x4_f32(false, a, false, b, (short)0, c,
                                                 false, false);
}

// ---------------------------------------------------------------------------
// Generic (optionally controlled) 2x2 complex gate on the distributed state.
// Lane holds amplitudes idx = lane*8 + r, r = 0..7 (idx bit w <-> wire 7-w).
// tb = target bit, cb = control bit (-1 = none).
// ---------------------------------------------------------------------------
static __device__ void apply_c2x2(float2 a[8], int lane, int cb, int tb,
                                  float2 m00, float2 m01, float2 m10, float2 m11) {
    float2 f[8];
    if (tb >= 3) {
        int xl = 1 << (tb - 3);
#pragma unroll
        for (int r = 0; r < 8; ++r) f[r] = shflx2(a[r], xl);
    } else {
        int xr = 1 << tb;
        float2 tmp[8];
#pragma unroll
        for (int r = 0; r < 8; ++r) tmp[r] = a[r];
#pragma unroll
        for (int r = 0; r < 8; ++r) f[r] = tmp[r ^ xr];
    }
#pragma unroll
    for (int r = 0; r < 8; ++r) {
        int idx = lane * 8 + r;
        if (cb >= 0 && !((idx >> cb) & 1)) continue;
        float2 x = a[r], y = f[r];
        float2 u, v;
        if ((idx >> tb) & 1) { u = cmul(m10, y); v = cmul(m11, x); }
        else                 { u = cmul(m00, x); v = cmul(m01, y); }
        a[r] = make_float2(u.x + v.x, u.y + v.y);
    }
}

static __device__ void apply_cr(float2 a[8], int lane, int branch, int cb, int tb, float t) {
    float s, c;
    sincosf(0.5f * t, &s, &c);
    float2 m00, m01, m10, m11;
    if (branch == 0) {            // CRX
        m00 = make_float2(c, 0.f);  m11 = m00;
        m01 = make_float2(0.f, -s); m10 = m01;
    } else if (branch == 1) {     // CRY
        m00 = make_float2(c, 0.f);  m11 = m00;
        m01 = make_float2(-s, 0.f); m10 = make_float2(s, 0.f);
    } else {                      // CRZ
        m00 = make_float2(c, -s);   m11 = make_float2(c, s);
        m01 = make_float2(0.f, 0.f); m10 = m01;
    }
    apply_c2x2(a, lane, cb, tb, m00, m01, m10, m11);
}

static __device__ void apply_u3(float2 a[8], int lane, int tb, const float* p) {
    float t = p[0], ph = p[1], la = p[2];
    float st, ct;  sincosf(0.5f * t, &st, &ct);
    float sl, cl;  sincosf(la, &sl, &cl);
    float sp, cp;  sincosf(ph, &sp, &cp);
    float spl, cpl; sincosf(ph + la, &spl, &cpl);
    float2 m00 = make_float2(ct, 0.f);
    float2 m01 = make_float2(-cl * st, -sl * st);
    float2 m10 = make_float2(cp * st, sp * st);
    float2 m11 = make_float2(cpl * ct, spl * ct);
    apply_c2x2(a, lane, -1, tb, m00, m01, m10, m11);
}

// ---------------------------------------------------------------------------
// Setup: build the three 256x256 branch unitaries.
// Interleaved layout: W[branch][k][j][comp] = U[j][k].{re,im}
// (wave = one basis column k; float2 stores are fully coalesced)
// ---------------------------------------------------------------------------
__global__ __launch_bounds__(256)
void qcnn_build_U(const float* __restrict__ crw, const float* __restrict__ u3p,
                  float* __restrict__ W) {
    int lane = threadIdx.x & 31;
    int wv = threadIdx.x >> 5;
    int gw = blockIdx.x * 8 + wv;     // 0..767
    int branch = gw >> 8;
    int col = gw & 255;

    float2 a[8];
#pragma unroll
    for (int r = 0; r < 8; ++r) {
        int idx = lane * 8 + r;
        a[r] = make_float2(idx == col ? 1.f : 0.f, 0.f);
    }
    // CR layer A: pairs (0,1)(2,3)(4,5)(6,7)(1,2)(3,4)(5,6) -> (cb,tb)
    const int cbA[7] = {7, 5, 3, 1, 6, 4, 2};
    const int tbA[7] = {6, 4, 2, 0, 5, 3, 1};
    for (int i = 0; i < 7; ++i) apply_cr(a, lane, branch, cbA[i], tbA[i], crw[i]);
    // U3 on wires 1,3,5,7 -> target bits 6,4,2,0
    const int tbU[4] = {6, 4, 2, 0};
    for (int g = 0; g < 4; ++g) apply_u3(a, lane, tbU[g], u3p + branch * 18 + g * 3);
    // CR layer B: (1,3)(5,7)(3,5) with weights 7,8,9
    apply_cr(a, lane, branch, 6, 4, crw[7]);
    apply_cr(a, lane, branch, 2, 0, crw[8]);
    apply_cr(a, lane, branch, 4, 2, crw[9]);
    // U3 on wires 3,7 with params rows 4,5
    apply_u3(a, lane, 4, u3p + branch * 18 + 12);
    apply_u3(a, lane, 0, u3p + branch * 18 + 15);

    float2* Wri = (float2*)W + (size_t)branch * 65536;
#pragma unroll
    for (int r = 0; r < 8; ++r) {
        int j = lane * 8 + r;
        Wri[col * 256 + j] = a[r];      // (re, im) interleaved
    }
}

// ---------------------------------------------------------------------------
// Main: 1 wave = 1 sample sim (4 cycles), then 16-wave cooperative complex
// GEMM (f32 WMMA) against the 3 branch unitaries + feature reduce + MLP.
// ---------------------------------------------------------------------------
#define LDSTRIDE 260   // 256 + pad, avoids 16-way LDS bank conflicts

static __device__ __forceinline__ void rxx(float2 a[8], int xl, int xr, float c, float s) {
    float2 f[8];
    if (xr) {
        float2 tmp[8];
#pragma unroll
        for (int r = 0; r < 8; ++r) tmp[r] = a[r];
#pragma unroll
        for (int r = 0; r < 8; ++r) f[r] = tmp[r ^ xr];
    } else {
#pragma unroll
        for (int r = 0; r < 8; ++r) f[r] = a[r];
    }
    if (xl) {
#pragma unroll
        for (int r = 0; r < 8; ++r) f[r] = shflx2(f[r], xl);
    }
#pragma unroll
    for (int r = 0; r < 8; ++r)   // a' = c*a - i*s*f
        a[r] = make_float2(c * a[r].x + s * f[r].y, c * a[r].y - s * f[r].x);
}

__global__ __launch_bounds__(512)
void qcnn_main(const float* __restrict__ theta, const float* __restrict__ phi,
               const float* __restrict__ W,
               const float* __restrict__ W1, const float* __restrict__ b1,
               const float* __restrict__ W2, const float* __restrict__ b2,
               float* __restrict__ out, int n) {
    __shared__ float Are[16 * LDSTRIDE];
    __shared__ float Aim[16 * LDSTRIDE];
    __shared__ float feat[16 * 6];

    int tid = threadIdx.x;
    int lane = tid & 31, wv = tid >> 5;      // wv = local sample (0..15)
    if (tid < 96) feat[tid] = 0.f;

    int sample = blockIdx.x * 16 + wv;
    int sc = sample < n ? sample : n - 1;
    float th = theta[sc], ph = phi[sc];

    // -------- per-sample statevector simulation (4 cycles) --------
    float2 a[8], pz[8];
#pragma unroll
    for (int r = 0; r < 8; ++r) {
        int idx = lane * 8 + r;
        a[r] = make_float2(0.0625f, 0.f);                 // H^8 |0>
        float ang = ph * (float)(2 * __popc(idx) - 8);    // fused 8x RZ
        float sv, cv; sincosf(ang, &sv, &cv);
        pz[r] = make_float2(cv, sv);
    }
    float st_, ct_;
    sincosf(th, &st_, &ct_);
    for (int cyc = 0; cyc < 4; ++cyc) {
#pragma unroll
        for (int r = 0; r < 8; ++r) a[r] = cmul(a[r], pz[r]);
        rxx(a, 24, 0, ct_, st_);   // (0,1): idx bits 7,6
        rxx(a, 6, 0, ct_, st_);    // (2,3): bits 5,4
        rxx(a, 1, 4, ct_, st_);    // (4,5): bits 3,2
        rxx(a, 0, 3, ct_, st_);    // (6,7): bits 1,0
        rxx(a, 12, 0, ct_, st_);   // (1,2): bits 6,5
        rxx(a, 3, 0, ct_, st_);    // (3,4): bits 4,3
        rxx(a, 0, 6, ct_, st_);    // (5,6): bits 2,1
    }
#pragma unroll
    for (int r = 0; r < 8; ++r) {
        int idx = lane * 8 + r;
        Are[wv * LDSTRIDE + idx] = a[r].x;
        Aim[wv * LDSTRIDE + idx] = a[r].y;
    }
    __syncthreads();

    // -------- WMMA complex GEMM: t = U s, per branch, wave = one n-tile ----
    int mrow = lane & 15;                 // A row (sample) / B column within tile
    int khalf = (lane >> 4) << 1;         // K sub-offset per fragment layout
    int j = (wv << 4) + mrow;             // output amplitude index for B loads
    float z7 = (lane & 1) ? -1.f : 1.f;   // <Z_7>: idx bit 0
    float z3 = (wv & 1) ? -1.f : 1.f;     // <Z_3>: idx bit 4 (uniform per n-tile)

    for (int br = 0; br < 3; ++br) {
        const float2* Wri = (const float2*)W + (size_t)br * 65536;
        const float* are = &Are[mrow * LDSTRIDE + khalf];
        const float* aim = &Aim[mrow * LDSTRIDE + khalf];
        const float2* wrow = Wri + (size_t)khalf * 256 + j;
        v8f P = {}, Q = {}, R = {}, S = {};
        for (int k0 = 0; k0 < 256; k0 += 4) {
            v2f ar; ar[0] = are[k0]; ar[1] = are[k0 + 1];
            v2f ai; ai[0] = aim[k0]; ai[1] = aim[k0 + 1];
            float2 l0 = wrow[(size_t)k0 * 256];          // (re,im) at row k
            float2 l1 = wrow[(size_t)(k0 + 1) * 256];    // (re,im) at row k+1
            v2f brr; brr[0] = l0.x; brr[1] = l1.x;
            v2f bii; bii[0] = l0.y; bii[1] = l1.y;
            P = wmma_f32(ar, brr, P);   // Re*Re
            Q = wmma_f32(ai, bii, Q);   // Im*Im
            R = wmma_f32(ar, bii, R);   // Re*Im
            S = wmma_f32(ai, brr, S);   // Im*Re
        }
#pragma unroll
        for (int v = 0; v < 8; ++v) {
            float tr = P[v] - Q[v];
            float ti = R[v] + S[v];
            float mag = tr * tr + ti * ti;
            float s3 = mag;
            float s7 = mag * z7;
#pragma unroll
            for (int off = 1; off < 16; off <<= 1) {
                s3 += __shfl_xor(s3, off, 32);
                s7 += __shfl_xor(s7, off, 32);
            }
            if (mrow == 0) {
                int m = v + ((lane >> 4) << 3);   // C-layout: M = v + 8*(lane>=16)
                atomicAdd(&feat[m * 6 + br * 2 + 0], s3 * z3);
                atomicAdd(&feat[m * 6 + br * 2 + 1], s7);
            }
        }
    }
    __syncthreads();

    // -------- tiny MLP head: tanh(feat @ W1^T + b1) @ W2^T + b2, sigmoid ---
    if (tid < 16) {
        int s = blockIdx.x * 16 + tid;
        if (s < n) {
            float acc = b2[0];
#pragma unroll
            for (int jj = 0; jj < 12; ++jj) {
                float hv = b1[jj];
#pragma unroll
                for (int i = 0; i < 6; ++i) hv += W1[jj * 6 + i] * feat[tid * 6 + i];
                acc += W2[jj] * tanhf(hv);
            }
            out[s] = 1.f / (1.f + expf(-acc));
        }
    }
}

extern "C" void kernel_launch(void* const* d_in, const int* in_sizes, int n_in,
                              void* d_out, int out_size, void* d_ws, size_t ws_size,
                              hipStream_t stream) {
    const float* theta = (const float*)d_in[0];
    const float* phi   = (const float*)d_in[1];
    const float* crw   = (const float*)d_in[2];
    const float* u3p   = (const float*)d_in[3];
    const float* W1    = (const float*)d_in[4];
    const float* b1    = (const float*)d_in[5];
    const float* W2    = (const float*)d_in[6];
    const float* b2    = (const float*)d_in[7];

    float* W = (float*)d_ws;                  // 3*2*256*256 floats = 1.5 MB
    int n = in_sizes[0];

    qcnn_build_U<<<96, 256, 0, stream>>>(crw, u3p, W);
    int blocks = (n + 15) / 16;
    qcnn_main<<<blocks, 512, 0, stream>>>(theta, phi, W, W1, b1, W2, b2,
                                          (float*)d_out, n);
}